// HitNet_90117003805323
// MI455X (gfx1250) — compile-verified
//
#include <hip/hip_runtime.h>
#include <cstdint>

#define NRAYS    32768
#define NPTS     200
#define HASHMASK ((1u << 19) - 1u)

// Trilinear interp from a dense R^3 table (idx = x + y*R + z*R^2).
// Works for both LDS-resident and global tables (address space inferred).
__device__ __forceinline__ float enc_dense(const float* __restrict__ tab,
                                           float X, float Y, float Z, int R) {
    const int   Rm1 = R - 1;
    const float s   = (float)Rm1;
    float px = X * s, py = Y * s, pz = Z * s;
    float fx = floorf(px), fy = floorf(py), fz = floorf(pz);
    int ix = min((int)fx, Rm1), iy = min((int)fy, Rm1), iz = min((int)fz, Rm1);
    float tx = px - fx, ty = py - fy, tz = pz - fz;
    float sx = 1.0f - tx, sy = 1.0f - ty, sz = 1.0f - tz;
    int ix1 = min(ix + 1, Rm1), iy1 = min(iy + 1, Rm1), iz1 = min(iz + 1, Rm1);
    int r0 = iy * R, r1 = iy1 * R;
    int p0 = iz * R * R, p1 = iz1 * R * R;
    float c000 = tab[ix  + r0 + p0], c100 = tab[ix1 + r0 + p0];
    float c010 = tab[ix  + r1 + p0], c110 = tab[ix1 + r1 + p0];
    float c001 = tab[ix  + r0 + p1], c101 = tab[ix1 + r0 + p1];
    float c011 = tab[ix  + r1 + p1], c111 = tab[ix1 + r1 + p1];
    float lo = sz * (sy * (sx * c000 + tx * c100) + ty * (sx * c010 + tx * c110));
    float hi = tz * (sy * (sx * c001 + tx * c101) + ty * (sx * c011 + tx * c111));
    return lo + hi;
}

__device__ __forceinline__ uint32_t hash3(uint32_t a, uint32_t b, uint32_t c) {
    return (a ^ (b * 2654435761u) ^ (c * 805459861u)) & HASHMASK;
}

// Trilinear interp from the hashed level (R = 128).
__device__ __forceinline__ float enc_hash(const float* __restrict__ tab,
                                          float X, float Y, float Z, int R) {
    const int   Rm1 = R - 1;
    const float s   = (float)Rm1;
    float px = X * s, py = Y * s, pz = Z * s;
    float fx = floorf(px), fy = floorf(py), fz = floorf(pz);
    uint32_t ix = (uint32_t)min((int)fx, Rm1), iy = (uint32_t)min((int)fy, Rm1),
             iz = (uint32_t)min((int)fz, Rm1);
    float tx = px - fx, ty = py - fy, tz = pz - fz;
    float sx = 1.0f - tx, sy = 1.0f - ty, sz = 1.0f - tz;
    uint32_t ix1 = min(ix + 1u, (uint32_t)Rm1), iy1 = min(iy + 1u, (uint32_t)Rm1),
             iz1 = min(iz + 1u, (uint32_t)Rm1);
    float c000 = tab[hash3(ix,  iy,  iz )], c100 = tab[hash3(ix1, iy,  iz )];
    float c010 = tab[hash3(ix,  iy1, iz )], c110 = tab[hash3(ix1, iy1, iz )];
    float c001 = tab[hash3(ix,  iy,  iz1)], c101 = tab[hash3(ix1, iy,  iz1)];
    float c011 = tab[hash3(ix,  iy1, iz1)], c111 = tab[hash3(ix1, iy1, iz1)];
    float lo = sz * (sy * (sx * c000 + tx * c100) + ty * (sx * c010 + tx * c110));
    float hi = tz * (sy * (sx * c001 + tx * c101) + ty * (sx * c011 + tx * c111));
    return lo + hi;
}

// One wave (32 lanes) per ray; 8 rays per 256-thread block.
__global__ __launch_bounds__(256) void hitnet_kernel(
    const float* __restrict__ x,
    const float* __restrict__ emb0, const float* __restrict__ emb1,
    const float* __restrict__ emb2, const float* __restrict__ emb3,
    const float* __restrict__ emb4,
    const float* __restrict__ w_lin, const float* __restrict__ b_lin,
    float* __restrict__ out_hits, float* __restrict__ out_full,
    float* __restrict__ out_idx) {
    // LDS cache for the two hottest tables: emb0 (512 f32) + emb1 (4096 f32) = 18KB.
    __shared__ float smem[512 + 4096];

    const int tid = threadIdx.x;

    // Stage via CDNA5 async global->LDS DMA path (ASYNCcnt-tracked).
    // LDS offset derived from the actual address (AS3->flat cast: low 32 bits
    // are the LDS byte offset). The ptrtoint of &smem[...] fed into the asm
    // captures the LDS array's address, so the "memory"-clobbering asm is a
    // potential writer of smem and its reads cannot be folded to poison.
    for (int i = tid; i < (512 + 4096) / 4; i += 256) {   // 1152 x b128
        unsigned lds_off = (unsigned)(uintptr_t)(&smem[i * 4]);
        const float* g = (i < 128) ? (emb0 + i * 4) : (emb1 + (i - 128) * 4);
        unsigned long long ga = (unsigned long long)g;
        asm volatile("global_load_async_to_lds_b128 %0, %1, off"
                     :: "v"(lds_off), "v"(ga)
                     : "memory");
    }
    asm volatile("s_wait_asynccnt 0x0" ::: "memory");
    __syncthreads();

    const int lane = tid & 31;
    const int ray  = blockIdx.x * 8 + (tid >> 5);

    const float4 r = ((const float4*)x)[ray];   // uv.x, uv.y, st.x, st.y
    const float w0 = w_lin[0], w1 = w_lin[1], w2 = w_lin[2],
                w3 = w_lin[3], w4 = w_lin[4];
    const float bl = b_lin[0];

    float vmax     = -3.0e38f;
    int   firstIdx = 0x7FFFFFFF;

    for (int i = lane; i < NPTS; i += 32) {
        float t   = (float)i * (1.0f / 199.0f);
        float omt = 1.0f - t;
        float X = r.x * omt + r.z * t;
        float Y = r.y * omt + r.w * t;
        float Z = t;

        float f0 = enc_dense(smem,        X, Y, Z, 8);    // LDS gathers
        float f1 = enc_dense(smem + 512,  X, Y, Z, 16);   // LDS gathers
        float f2 = enc_dense(emb2,        X, Y, Z, 32);   // L0/L2 gathers
        float f3 = enc_dense(emb3,        X, Y, Z, 64);
        float f4 = enc_hash(emb4,         X, Y, Z, 128);

        float v   = fmaf(w0, f0, fmaf(w1, f1, fmaf(w2, f2, fmaf(w3, f3, fmaf(w4, f4, bl)))));
        float sig = 1.0f / (1.0f + __expf(-v));
        out_full[ray * NPTS + i] = sig;                   // coalesced per wave

        vmax = fmaxf(vmax, v);                            // sigmoid monotonic
        if (v > 0.0f) firstIdx = min(firstIdx, i);        // sig>0.5 <=> v>0
    }

    // wave32 reductions
    #pragma unroll
    for (int m = 16; m >= 1; m >>= 1) {
        vmax     = fmaxf(vmax, __shfl_xor(vmax, m, 32));
        firstIdx = min(firstIdx, __shfl_xor(firstIdx, m, 32));
    }

    if (lane == 0) {
        out_hits[ray] = 1.0f / (1.0f + __expf(-vmax));
        out_idx[ray]  = (firstIdx == 0x7FFFFFFF) ? 0.0f : (float)firstIdx;
    }
}

extern "C" void kernel_launch(void* const* d_in, const int* in_sizes, int n_in,
                              void* d_out, int out_size, void* d_ws, size_t ws_size,
                              hipStream_t stream) {
    (void)in_sizes; (void)n_in; (void)d_ws; (void)ws_size; (void)out_size;
    const float* x    = (const float*)d_in[0];
    const float* emb0 = (const float*)d_in[1];
    const float* emb1 = (const float*)d_in[2];
    const float* emb2 = (const float*)d_in[3];
    const float* emb3 = (const float*)d_in[4];
    const float* emb4 = (const float*)d_in[5];
    const float* wlin = (const float*)d_in[6];
    const float* blin = (const float*)d_in[7];

    float* out      = (float*)d_out;
    float* out_hits = out;                        // [32768]
    float* out_full = out + NRAYS;                // [32768*200]
    float* out_idx  = out + NRAYS + NRAYS * NPTS; // [32768]

    hitnet_kernel<<<dim3(NRAYS / 8), dim3(256), 0, stream>>>(
        x, emb0, emb1, emb2, emb3, emb4, wlin, blin,
        out_hits, out_full, out_idx);
}